// GraphConv_33122787787043
// MI455X (gfx1250) — compile-verified
//
#include <hip/hip_runtime.h>

// GraphConv 2-hop scatter-mean + L2-normalize for MI455X (gfx1250, wave32).
//
// Design (see roofline analysis): entity table (25.6 MB) and agg buffer
// (25.6 MB) are L2-resident (192 MB L2), so the edge-parallel gather +
// global_atomic_add_f32 scatter is the right data path. WMMA f32 16x16x4
// computes 16 row-norms per wave in the normalize pass (A = squared tile
// fragment, B = all-ones -> every D element in row m is ||row m||^2).

typedef __attribute__((ext_vector_type(2))) float v2f;
typedef __attribute__((ext_vector_type(8))) float v8f;

#define CH 64
#define NDRUGS 2000
#define LDS_STRIDE 68       // 64 + 4 floats pad -> conflict-free strided reads
#define EDGES_PER_WAVE 16   // amortize 4KB rel-table LDS preload per block

// ---------------------------------------------------------------- counts ----
__global__ void count_edges_kernel(const int* __restrict__ head,
                                   unsigned* __restrict__ counts, int E) {
    int e = blockIdx.x * blockDim.x + threadIdx.x;
    if (e < E) atomicAdd(&counts[head[e]], 1u);
}

// ------------------------------------------------------------- rel norms ----
// rel_res = rel0 + l2norm(rel0) + l2norm(l2norm(rel0)); relN = l2norm(rel0)
// (hop-1 scatter uses relN). 16 rows x 64 ch: trivial.
__global__ void relnorm_kernel(const float* __restrict__ rel0,
                               float* __restrict__ relN,
                               float* __restrict__ relRes) {
    int r = threadIdx.x;
    if (r >= 16) return;
    float v0[CH], v1[CH];
    float s0 = 0.f;
    for (int c = 0; c < CH; ++c) { float x = rel0[r * CH + c]; v0[c] = x; s0 += x * x; }
    float inv0 = 1.0f / fmaxf(sqrtf(s0), 1e-12f);
    float s1 = 0.f;
    for (int c = 0; c < CH; ++c) { float x = v0[c] * inv0; v1[c] = x; relN[r * CH + c] = x; s1 += x * x; }
    float inv1 = 1.0f / fmaxf(sqrtf(s1), 1e-12f);
    for (int c = 0; c < CH; ++c) relRes[r * CH + c] = v0[c] + v1[c] + v1[c] * inv1;
}

// --------------------------------------------------------------- scatter ----
// One wave per edge: 64 ch / 32 lanes = one float2/lane; gather from L2-
// resident ent table, scale by LDS rel row, 2x global_atomic_add_f32.
__global__ void scatter_kernel(const int* __restrict__ head,
                               const int* __restrict__ tail,
                               const int* __restrict__ etype,
                               const float* __restrict__ ent,
                               const float* __restrict__ rel,
                               float* __restrict__ agg, int E) {
    __shared__ float srel[16 * CH];                 // 4 KB relation table
    ((float4*)srel)[threadIdx.x] = ((const float4*)rel)[threadIdx.x]; // 256*16B
    __syncthreads();

    int wave = threadIdx.x >> 5;
    int lane = threadIdx.x & 31;
    int base = (blockIdx.x * 8 + wave) * EDGES_PER_WAVE;
#pragma unroll 4
    for (int i = 0; i < EDGES_PER_WAVE; ++i) {
        int e = base + i;
        if (e < E) {
            int h = head[e], t = tail[e], r = etype[e];
            float2 ev = ((const float2*)(ent + (size_t)t * CH))[lane];
            float2 rv = ((const float2*)(srel + r * CH))[lane];
            float* ap = agg + (size_t)h * CH + lane * 2;
            atomicAdd(ap + 0, ev.x * rv.x);
            atomicAdd(ap + 1, ev.y * rv.y);
        }
    }
}

// ------------------------------------------------------------- normalize ----
// One wave per 16-row tile. Phase 1: coalesced 16x64 load, apply 1/denom,
// stage into padded LDS. Phase 2: 16x V_WMMA_F32_16X16X4_F32 accumulating
// D = (A^2) x Ones -> lanes 0-15 hold ||row 0..7||^2 in c[0..7], lanes 16-31
// hold rows 8..15. Phase 3: broadcast-shuffle the needed norm, scale, write
// entOut and accumulate ent_res (+ drug_res for rows < 2000).
__global__ void normalize_kernel(const float* __restrict__ agg,
                                 const unsigned* __restrict__ counts,
                                 float* __restrict__ entOut,
                                 float* __restrict__ entRes,
                                 float* __restrict__ drugRes, int nTiles) {
    __shared__ float smem[8 * 16 * LDS_STRIDE];     // 34.8 KB / block (8 waves)
    int wave = threadIdx.x >> 5;
    int lane = threadIdx.x & 31;
    int tile = blockIdx.x * 8 + wave;
    if (tile >= nTiles) return;  // wave-uniform; no block barriers below -> safe

    float* T = smem + wave * (16 * LDS_STRIDE);
    int p   = lane >> 4;      // half-wave id (A/C fragment K/M split)
    int l15 = lane & 15;
    int tileBase = tile * 16;

    // Phase 1: load + 1/denom + stage (row stride 68 -> conflict-free reads)
#pragma unroll
    for (int k = 0; k < 8; ++k) {
        int row = 2 * k + p;
        int col = l15 * 4;
        int gr  = tileBase + row;
        float4 v = *(const float4*)(agg + (size_t)gr * CH + col);
        float invd = 1.0f / fmaxf((float)counts[gr], 1.0f);
        *(float4*)&T[row * LDS_STRIDE + col] =
            make_float4(v.x * invd, v.y * invd, v.z * invd, v.w * invd);
    }
    __builtin_amdgcn_wave_barrier();

    // Phase 2: norms via WMMA. A fragment (lane l: M=l&15, K split by half-wave)
    v8f c = {0.f, 0.f, 0.f, 0.f, 0.f, 0.f, 0.f, 0.f};
    v2f b; b.x = 1.0f; b.y = 1.0f;   // all-ones B: layout-independent result
#pragma unroll
    for (int kc = 0; kc < 16; ++kc) {
        v2f v = *(const v2f*)&T[l15 * LDS_STRIDE + kc * 4 + 2 * p];
        v2f a; a.x = v.x * v.x; a.y = v.y * v.y;
        c = __builtin_amdgcn_wmma_f32_16x16x4_f32(false, a, false, b,
                                                  (short)0, c, false, false);
    }

    // Phase 3: scale + write. Lane handles rows {2k+p}. Norm of row m lives in
    // c[m] on lanes 0-15 (m<8) or c[m-8] on lanes 16-31 (m>=8); broadcast it.
#pragma unroll
    for (int k = 0; k < 8; ++k) {
        float evenV = (k < 4) ? c[2 * k]     : __shfl(c[2 * k - 8], 16, 32);
        float oddV  = (k < 4) ? __shfl(c[2 * k + 1], 0, 32) : c[2 * k - 7];
        float n2  = p ? oddV : evenV;
        float inv = 1.0f / fmaxf(sqrtf(n2), 1e-12f);
        int row = 2 * k + p;
        int col = l15 * 4;
        int gr  = tileBase + row;
        float4 w = *(const float4*)&T[row * LDS_STRIDE + col];
        float4 o = make_float4(w.x * inv, w.y * inv, w.z * inv, w.w * inv);
        *(float4*)(entOut + (size_t)gr * CH + col) = o;
        float4* er = (float4*)(entRes + (size_t)gr * CH + col);
        float4 rv = *er;
        *er = make_float4(rv.x + o.x, rv.y + o.y, rv.z + o.z, rv.w + o.w);
        if (gr < NDRUGS) {
            float4* dr = (float4*)(drugRes + (size_t)gr * CH + col);
            float4 dv = *dr;
            *dr = make_float4(dv.x + o.x, dv.y + o.y, dv.z + o.z, dv.w + o.w);
        }
    }
}

// ----------------------------------------------------------------- launch ---
extern "C" void kernel_launch(void* const* d_in, const int* in_sizes, int n_in,
                              void* d_out, int out_size, void* d_ws, size_t ws_size,
                              hipStream_t stream) {
    const float* drug_emb     = (const float*)d_in[0];
    const float* entity_emb   = (const float*)d_in[1];
    const float* relation_emb = (const float*)d_in[2];
    const int*   edge_index   = (const int*)d_in[3];
    const int*   edge_type    = (const int*)d_in[4];

    int E     = in_sizes[4];          // 1,250,000
    int nEnt  = in_sizes[1] / CH;     // 100,000
    int nDrug = in_sizes[0] / CH;     // 2,000

    // Workspace layout (~51.7 MB): agg | entWs | counts | relN
    size_t entBytes = (size_t)nEnt * CH * sizeof(float);
    float*    agg    = (float*)d_ws;
    float*    entWs  = (float*)((char*)d_ws + entBytes);
    unsigned* counts = (unsigned*)((char*)d_ws + 2 * entBytes);
    float*    relN   = (float*)((char*)d_ws + 2 * entBytes + (size_t)nEnt * sizeof(unsigned));

    float* entRes  = (float*)d_out;
    float* drugRes = entRes + (size_t)nEnt * CH;
    float* relRes  = drugRes + (size_t)nDrug * CH;

    const int* head = edge_index;
    const int* tail = edge_index + E;

    // Residual init + per-head in-degree (hop-invariant)
    hipMemcpyAsync(entRes, entity_emb, entBytes, hipMemcpyDeviceToDevice, stream);
    hipMemcpyAsync(drugRes, drug_emb, (size_t)nDrug * CH * sizeof(float),
                   hipMemcpyDeviceToDevice, stream);
    hipMemsetAsync(counts, 0, (size_t)nEnt * sizeof(unsigned), stream);
    count_edges_kernel<<<(E + 255) / 256, 256, 0, stream>>>(head, counts, E);
    relnorm_kernel<<<1, 32, 0, stream>>>(relation_emb, relN, relRes);

    int nTiles        = nEnt / 16;                          // 6250
    int scatterBlocks = (E + 8 * EDGES_PER_WAVE - 1) / (8 * EDGES_PER_WAVE);
    int normBlocks    = (nTiles + 7) / 8;

    // Hop 0: gather from original embeddings
    hipMemsetAsync(agg, 0, entBytes, stream);
    scatter_kernel<<<scatterBlocks, 256, 0, stream>>>(head, tail, edge_type,
                                                      entity_emb, relation_emb, agg, E);
    normalize_kernel<<<normBlocks, 256, 0, stream>>>(agg, counts, entWs,
                                                     entRes, drugRes, nTiles);
    // Hop 1: gather from normalized hop-0 result, normalized relations
    hipMemsetAsync(agg, 0, entBytes, stream);
    scatter_kernel<<<scatterBlocks, 256, 0, stream>>>(head, tail, edge_type,
                                                      entWs, relN, agg, E);
    normalize_kernel<<<normBlocks, 256, 0, stream>>>(agg, counts, entWs,
                                                     entRes, drugRes, nTiles);
}